// RecurrentFormulationNetv2_21784074126006
// MI455X (gfx1250) — compile-verified
//
#include <hip/hip_runtime.h>

typedef __attribute__((ext_vector_type(2))) float v2f;
typedef __attribute__((ext_vector_type(8))) float v8f;

#define GEMM_BLOCK 256
#define TILE_ROWS  128
#define CHUNK      32
#define ALD        36    // Alds row stride (pad: conflict-free for A-frag reads)
#define WLD        136   // Wlds row stride (pad: conflict-free for B-frag reads)

// ---------------------------------------------------------------------------
// WMMA GEMM: out[N x ncols] = A[N x K] * W[K x ncols]
// A is read from up to 3 concatenated column segments (k<k1 -> s0, k<k2 -> s1,
// else s2), each with its own leading dim and optional fused ReLU.
// Uses V_WMMA_F32_16X16X4_F32 (f32 matrix pipe, exact f32 math).
// ---------------------------------------------------------------------------
template <int NT>
__global__ __launch_bounds__(GEMM_BLOCK)
void gemm_wmma_kernel(const float* __restrict__ s0, int ld0, int k1, int relu0,
                      const float* __restrict__ s1, int ld1, int k2, int relu1,
                      const float* __restrict__ s2, int ld2, int relu2,
                      const float* __restrict__ W, int K, int ncols,
                      float* __restrict__ out, int ldout, int N)
{
    __shared__ float Alds[TILE_ROWS * ALD];
    __shared__ float Wlds[CHUNK * WLD];

    const int tid  = threadIdx.x;
    const int wave = tid >> 5;
    const int lane = tid & 31;
    const int half = lane >> 4;   // 0: lanes 0-15, 1: lanes 16-31
    const int l16  = lane & 15;
    const int row0 = blockIdx.x * TILE_ROWS;

    v8f acc[NT];
#pragma unroll
    for (int i = 0; i < NT; ++i)
#pragma unroll
        for (int j = 0; j < 8; ++j) acc[i][j] = 0.0f;

    for (int k0 = 0; k0 < K; k0 += CHUNK) {
        // ---- stage W chunk into LDS (zero-fill then load valid region) ----
        for (int idx = tid; idx < CHUNK * WLD; idx += GEMM_BLOCK) Wlds[idx] = 0.0f;
        __syncthreads();
        for (int idx = tid; idx < CHUNK * ncols; idx += GEMM_BLOCK) {
            int kk = idx / ncols, n = idx - kk * ncols;
            int k  = k0 + kk;
            Wlds[kk * WLD + n] = (k < K) ? W[(size_t)k * ncols + n] : 0.0f;
        }
        // ---- stage A chunk (segmented sources, fused ReLU) ----
        for (int idx = tid; idx < TILE_ROWS * CHUNK; idx += GEMM_BLOCK) {
            int r = idx >> 5, kk = idx & (CHUNK - 1);
            int row = row0 + r, k = k0 + kk;
            float v = 0.0f;
            if (row < N && k < K) {
                if (k < k1)      { v = s0[(size_t)row * ld0 + k];        if (relu0) v = fmaxf(v, 0.0f); }
                else if (k < k2) { v = s1[(size_t)row * ld1 + (k - k1)]; if (relu1) v = fmaxf(v, 0.0f); }
                else             { v = s2[(size_t)row * ld2 + (k - k2)]; if (relu2) v = fmaxf(v, 0.0f); }
            }
            Alds[r * ALD + kk] = v;
        }
        __syncthreads();

        // ---- WMMA accumulate: 16x16 tiles, K step 4 ----
        const int abase = (wave * 16 + l16) * ALD + 2 * half;
#pragma unroll
        for (int kk = 0; kk < CHUNK; kk += 4) {
            v2f a;
            a.x = Alds[abase + kk];        // A[row][k0+kk+2*half]
            a.y = Alds[abase + kk + 1];    // A[row][k0+kk+2*half+1]
#pragma unroll
            for (int nt = 0; nt < NT; ++nt) {
                v2f b;
                b.x = Wlds[(kk + 2 * half)     * WLD + nt * 16 + l16];
                b.y = Wlds[(kk + 2 * half + 1) * WLD + nt * 16 + l16];
                acc[nt] = __builtin_amdgcn_wmma_f32_16x16x4_f32(
                    false, a, false, b, (short)0, acc[nt], false, false);
            }
        }
        __syncthreads();
    }

    // ---- store C: VGPR j -> row m0 + j + 8*half, col = nt*16 + l16 ----
    const int m0 = row0 + wave * 16;
#pragma unroll
    for (int nt = 0; nt < NT; ++nt) {
        int col = nt * 16 + l16;
        if (col < ncols) {
#pragma unroll
            for (int j = 0; j < 8; ++j) {
                int row = m0 + j + 8 * half;
                if (row < N) out[(size_t)row * ldout + col] = acc[nt][j];
            }
        }
    }
}

// ---------------------------------------------------------------------------
// Elementwise / graph kernels
// ---------------------------------------------------------------------------
__global__ void zero_kernel(float* p, int n) {
    int i = blockIdx.x * blockDim.x + threadIdx.x;
    if (i < n) p[i] = 0.0f;
}

__global__ void deg_init_kernel(float* deg_w, float* deg_1, int N) {
    int i = blockIdx.x * blockDim.x + threadIdx.x;
    if (i < N) { deg_w[i] = 1.0f; deg_1[i] = 1.0f; }   // +1 self loop
}

__global__ void deg_acc_kernel(const int* __restrict__ dst, const float* __restrict__ ew,
                               float* deg_w, float* deg_1, int E) {
    int e = blockIdx.x * blockDim.x + threadIdx.x;
    if (e < E) {
        int d = dst[e];
        atomicAdd(&deg_w[d], ew[e]);
        atomicAdd(&deg_1[d], 1.0f);
    }
}

__global__ void dinv_kernel(const float* __restrict__ deg_w, const float* __restrict__ deg_1,
                            float* dinv_w, float* inv_w, float* dinv_1, float* inv_1, int N) {
    int i = blockIdx.x * blockDim.x + threadIdx.x;
    if (i < N) {
        float dw = deg_w[i], d1 = deg_1[i];
        dinv_w[i] = rsqrtf(dw); inv_w[i] = 1.0f / dw;
        dinv_1[i] = rsqrtf(d1); inv_1[i] = 1.0f / d1;
    }
}

__global__ void norm_kernel(const int* __restrict__ src, const int* __restrict__ dst,
                            const float* __restrict__ ew,
                            const float* __restrict__ dinv_w, const float* __restrict__ dinv_1,
                            float* norm_w, float* norm_1, int E) {
    int e = blockIdx.x * blockDim.x + threadIdx.x;
    if (e < E) {
        int s = src[e], d = dst[e];
        norm_w[e] = dinv_w[s] * dinv_w[d] * ew[e];
        norm_1[e] = dinv_1[s] * dinv_1[d];
    }
}

// out[n,c] = xw[n,c]*inv[n] + bias[c]   (self term + bias; scatter adds on top)
__global__ void init128_kernel(float* __restrict__ out, const float* __restrict__ xw,
                               const float* __restrict__ inv, const float* __restrict__ bias, int N) {
    int idx = blockIdx.x * blockDim.x + threadIdx.x;
    if (idx < N * 128) {
        int n = idx >> 7, c = idx & 127;
        out[idx] = xw[idx] * inv[n] + bias[c];
    }
}

__global__ void scatter128_kernel(const float* __restrict__ norm, const int* __restrict__ src,
                                  const int* __restrict__ dst, const float* __restrict__ xw,
                                  float* out, int E) {
    int idx = blockIdx.x * blockDim.x + threadIdx.x;
    if (idx < E * 32) {
        int e = idx >> 5, g = idx & 31;
        float nr = norm[e];
        int s = src[e], d = dst[e];
        const float4 v = *reinterpret_cast<const float4*>(xw + (size_t)s * 128 + g * 4);
        float* o = out + (size_t)d * 128 + g * 4;
        atomicAdd(o + 0, nr * v.x);
        atomicAdd(o + 1, nr * v.y);
        atomicAdd(o + 2, nr * v.z);
        atomicAdd(o + 3, nr * v.w);
    }
}

__global__ void relu_kernel(float* p, int n) {
    int i = blockIdx.x * blockDim.x + threadIdx.x;
    if (i < n) p[i] = fmaxf(p[i], 0.0f);
}

__global__ void init8_kernel(float* __restrict__ out, int ld, const float* __restrict__ xw,
                             const float* __restrict__ inv, const float* __restrict__ bias, int N) {
    int idx = blockIdx.x * blockDim.x + threadIdx.x;
    if (idx < N * 8) {
        int n = idx >> 3, c = idx & 7;
        out[(size_t)n * ld + c] = xw[idx] * inv[n] + bias[c];
    }
}

__global__ void scatter8_kernel(const float* __restrict__ norm, const int* __restrict__ src,
                                const int* __restrict__ dst, const float* __restrict__ xw,
                                float* out, int ld, int E) {
    int idx = blockIdx.x * blockDim.x + threadIdx.x;
    if (idx < E * 8) {
        int e = idx >> 3, c = idx & 7;
        float nr = norm[e];
        int s = src[e], d = dst[e];
        atomicAdd(out + (size_t)d * ld + c, nr * xw[(size_t)s * 8 + c]);
    }
}

// ---------------------------------------------------------------------------
static inline dim3 g1(long n, int b = 256) { return dim3((unsigned)((n + b - 1) / b)); }

extern "C" void kernel_launch(void* const* d_in, const int* in_sizes, int n_in,
                              void* d_out, int out_size, void* d_ws, size_t ws_size,
                              hipStream_t stream)
{
    const float* F0        = (const float*)d_in[0];   // [N,8]
    const float* mesh_node = (const float*)d_in[1];   // [N,8]
    const float* mesh_edge = (const float*)d_in[2];   // [E]
    const int*   eidx      = (const int*)  d_in[3];   // [2,E]
    // d_in[4] = n_time scalar (device); derived from out_size instead.
    const float* W_in  = (const float*)d_in[5];       // [144,128]
    const float* b_in  = (const float*)d_in[6];       // [128]
    const float* net_W = (const float*)d_in[7];       // [3,128,128]
    const float* net_b = (const float*)d_in[8];       // [3,128]
    const float* W_out = (const float*)d_in[9];       // [256,8]
    const float* b_out = (const float*)d_in[10];      // [8]

    const int N = in_sizes[0] / 8;
    const int E = in_sizes[2];
    const int n_time = out_size / (N * 8);
    const int ldF = n_time * 8;
    const int* src = eidx;
    const int* dst = eidx + E;
    float* out = (float*)d_out;

    // workspace layout
    float* ws = (float*)d_ws;
    size_t off = 0;
    auto alloc = [&](size_t n) { float* p = ws + off; off += n; return p; };
    float* deg_w  = alloc(N);
    float* deg_1  = alloc(N);
    float* dinv_w = alloc(N);
    float* inv_w  = alloc(N);
    float* dinv_1 = alloc(N);
    float* inv_1  = alloc(N);
    float* norm_w = alloc(E);
    float* norm_1 = alloc(E);
    float* xw     = alloc((size_t)N * 128);
    float* h1     = alloc((size_t)N * 128);
    float* bufA   = alloc((size_t)N * 128);   // also holds h (pre-ReLU x)
    float* bufB   = alloc((size_t)N * 128);
    float* bufC   = alloc((size_t)N * 128);
    float* xw_out = alloc((size_t)N * 8);
    (void)ws_size;

    const int GB = (N + TILE_ROWS - 1) / TILE_ROWS;

    // ---- degree / normalization precompute (constant over timesteps) ----
    deg_init_kernel<<<g1(N), 256, 0, stream>>>(deg_w, deg_1, N);
    deg_acc_kernel<<<g1(E), 256, 0, stream>>>(dst, mesh_edge, deg_w, deg_1, E);
    dinv_kernel<<<g1(N), 256, 0, stream>>>(deg_w, deg_1, dinv_w, inv_w, dinv_1, inv_1, N);
    norm_kernel<<<g1(E), 256, 0, stream>>>(src, dst, mesh_edge, dinv_w, dinv_1, norm_w, norm_1, E);

    // h = 0 at t=0; h lives in bufA (pre-ReLU x from previous timestep)
    zero_kernel<<<g1((long)N * 128), 256, 0, stream>>>(bufA, N * 128);
    float* h = bufA;

    for (int t = 0; t < n_time; ++t) {
        const float* Fc  = (t == 0) ? F0 : (out + (size_t)(t - 1) * 8);
        const int    ldc = (t == 0) ? 8 : ldF;

        // h1 = relu(gcn_conv(concat[F_cur, mesh_node, h], W_in, b_in, mesh_edge))
        gemm_wmma_kernel<8><<<GB, GEMM_BLOCK, 0, stream>>>(
            Fc, ldc, 8, 0, mesh_node, 8, 16, 0, h, 128, 0,
            W_in, 144, 128, xw, 128, N);
        init128_kernel<<<g1((long)N * 128), 256, 0, stream>>>(h1, xw, inv_w, b_in, N);
        scatter128_kernel<<<g1((long)E * 32), 256, 0, stream>>>(norm_w, src, dst, xw, h1, E);
        relu_kernel<<<g1((long)N * 128), 256, 0, stream>>>(h1, N * 128);

        // 3 net layers with unit edge weights; x ends in bufA (pre-ReLU)
        const float* xin = h1;
        float* outs3[3] = { bufB, bufC, bufA };
        for (int i = 0; i < 3; ++i) {
            float* o = outs3[i];
            gemm_wmma_kernel<8><<<GB, GEMM_BLOCK, 0, stream>>>(
                xin, 128, 0, 0, xin, 128, 0, 0, xin, 128, 0,
                net_W + (size_t)i * 128 * 128, 128, 128, xw, 128, N);
            init128_kernel<<<g1((long)N * 128), 256, 0, stream>>>(o, xw, inv_1, net_b + (size_t)i * 128, N);
            scatter128_kernel<<<g1((long)E * 32), 256, 0, stream>>>(norm_1, src, dst, xw, o, E);
            if (i < 2) relu_kernel<<<g1((long)N * 128), 256, 0, stream>>>(o, N * 128);
            xin = o;
        }
        float* x = bufA;   // pre-ReLU final layer output (= h for next timestep)

        // F_next = gcn_conv(concat[h1, relu(x)], W_out, b_out, ones)
        gemm_wmma_kernel<1><<<GB, GEMM_BLOCK, 0, stream>>>(
            h1, 128, 128, 0, x, 128, 256, 1, x, 128, 0,
            W_out, 256, 8, xw_out, 8, N);
        init8_kernel<<<g1((long)N * 8), 256, 0, stream>>>(out + (size_t)t * 8, ldF, xw_out, inv_1, b_out, N);
        scatter8_kernel<<<g1((long)E * 8), 256, 0, stream>>>(norm_1, src, dst, xw_out, out + (size_t)t * 8, ldF, E);

        h = bufA;  // x (pre-ReLU) is next timestep's hidden state
    }
}